// MaskedGCN_65816078844665
// MI455X (gfx1250) — compile-verified
//
#include <hip/hip_runtime.h>

typedef __attribute__((ext_vector_type(2))) float v2f;
typedef __attribute__((ext_vector_type(8))) float v8f;

#define DNODE 128  // D_IN == D_OUT == 128

// ---------------------------------------------------------------------------
// fast fp32 global atomic add (global_atomic_add_f32, no CAS loop)
// ---------------------------------------------------------------------------
__device__ __forceinline__ void atomic_add_f32(float* p, float v) {
    unsafeAtomicAdd(p, v);
}

// ---------------------------------------------------------------------------
// degree: deg[n] = 1 (self loop) + #edges with dst == n
// ---------------------------------------------------------------------------
__global__ __launch_bounds__(256) void k_deg_init(float* __restrict__ deg, int n_nodes) {
    int i = blockIdx.x * 256 + threadIdx.x;
    if (i < n_nodes) deg[i] = 1.0f;
}

__global__ __launch_bounds__(256) void k_deg_count(const long long* __restrict__ edst,
                                                   float* __restrict__ deg, int n_edges) {
    int e = blockIdx.x * 256 + threadIdx.x;
    if (e < n_edges) atomic_add_f32(&deg[(int)edst[e]], 1.0f);
}

__global__ __launch_bounds__(256) void k_deg_to_dinv(float* __restrict__ deg, int n_nodes) {
    int i = blockIdx.x * 256 + threadIdx.x;
    if (i < n_nodes) deg[i] = rsqrtf(deg[i]);  // deg >= 1 always (self loop)
}

// ---------------------------------------------------------------------------
// pack W[k][c] (row-major) into WMMA B-fragment "pair layout":
//   pair p = k>>1 holds (W[2p][c], W[2p+1][c]) contiguously -> one ds_load_b64
//   per fragment. Odd pair-rows are rotated by 32 words so the two half-waves
//   (pair p / pair p+1) hit disjoint LDS bank halves.
// ---------------------------------------------------------------------------
__device__ __forceinline__ int wp_findex(int k, int c) {
    int p = k >> 1;
    int w = (((c << 1) + ((p & 1) << 5)) & 255) + (k & 1);
    return p * 256 + w;
}

__global__ __launch_bounds__(256) void k_pack_w(const float* __restrict__ W,
                                                float* __restrict__ Wp) {
    int i = blockIdx.x * 256 + threadIdx.x;  // 16384 elements
    if (i < DNODE * DNODE) {
        int k = i >> 7, c = i & 127;
        Wp[wp_findex(k, c)] = W[i];
    }
}

// ---------------------------------------------------------------------------
// h = (x * mask) @ W  via V_WMMA_F32_16X16X4_F32 (full fp32 precision)
// block = 128 threads = 4 waves; each wave: 16 rows x 128 cols of h.
// Wp (64KB, pair layout) is staged into LDS with global_load_async_to_lds_b128
// (ASYNCcnt path), then each B fragment is a single aligned ds_load_b64.
// ---------------------------------------------------------------------------
__global__ __launch_bounds__(128) void k_gemm_wmma(const float* __restrict__ x,
                                                   const float* __restrict__ Wp,
                                                   const int* __restrict__ mask,
                                                   float* __restrict__ h,
                                                   int n_nodes) {
    __shared__ float Wlds[DNODE * DNODE];  // 64 KB, pair layout

    const int tid  = threadIdx.x;
    const int lane = tid & 31;
    const int wave = tid >> 5;

    // async copy Wp -> LDS: 128 threads x 16B x 32 iters = 64KB
    {
        const unsigned lbase = (unsigned)(uintptr_t)&Wlds[0];
        const unsigned long long gbase = (unsigned long long)(uintptr_t)Wp;
#pragma unroll 4
        for (int it = 0; it < 32; ++it) {
            unsigned off = (unsigned)((it * 128 + tid) * 16);
            unsigned laddr = lbase + off;
            unsigned long long gaddr = gbase + off;
            asm volatile("global_load_async_to_lds_b128 %0, %1, off"
                         :: "v"(laddr), "v"(gaddr) : "memory");
        }
        asm volatile("s_wait_asynccnt 0x0" ::: "memory");
    }
    __syncthreads();

    const int row_base = (blockIdx.x * 4 + wave) * 16;
    if (row_base >= n_nodes) return;  // uniform per wave -> EXEC stays all-1s

    const int mrow   = lane & 15;        // M (and N) index covered by this lane
    const int half   = lane >> 4;        // 0: K=k0,k0+1   1: K=k0+2,k0+3
    const int kphase = half * 2;
    const int arow   = row_base + mrow;
    const bool rv    = arow < n_nodes;
    const float am   = rv ? (float)mask[arow] : 0.0f;  // 0/1 mask folded into A
    const float* xrow = x + (size_t)(rv ? arow : 0) * DNODE;

    // per-tile LDS word offset within a pair-row (loop-invariant over k)
    int wofs[8];
#pragma unroll
    for (int t = 0; t < 8; ++t)
        wofs[t] = (((t * 16 + mrow) << 1) + (half << 5)) & 255;
    const int pbase = half * 256;  // pair p = k0/2 + half

    v8f acc[8] = {};  // 8 N-tiles of 16 -> full 128 output cols

    for (int k0 = 0; k0 < DNODE; k0 += 4) {
        // A fragment: 16x4 tile of masked x (8-byte aligned float2 load)
        v2f a = *(const v2f*)(xrow + k0 + kphase);
        a.x *= am;
        a.y *= am;
#pragma unroll
        for (int t = 0; t < 8; ++t) {
            // B fragment: single aligned b64 load from pair-layout LDS
            v2f b = *(const v2f*)&Wlds[k0 * 128 + pbase + wofs[t]];
            acc[t] = __builtin_amdgcn_wmma_f32_16x16x4_f32(
                /*neg_a=*/false, a, /*neg_b=*/false, b,
                /*c_mod=*/(short)0, acc[t], /*reuse_a=*/false, /*reuse_b=*/false);
        }
    }

    // C/D layout: VGPR v -> M = v (lanes 0-15) or v+8 (lanes 16-31); N = lane&15
    const int roff = half * 8;
    float* hb = h + (size_t)row_base * DNODE;
    if (row_base + 16 <= n_nodes) {
        // full tile: unguarded stores (no per-store EXEC juggling)
#pragma unroll
        for (int t = 0; t < 8; ++t)
#pragma unroll
            for (int v = 0; v < 8; ++v)
                hb[(size_t)(v + roff) * DNODE + t * 16 + mrow] = acc[t][v];
    } else {
#pragma unroll
        for (int t = 0; t < 8; ++t)
#pragma unroll
            for (int v = 0; v < 8; ++v) {
                int r = row_base + v + roff;
                if (r < n_nodes) h[(size_t)r * DNODE + t * 16 + mrow] = acc[t][v];
            }
    }
}

// ---------------------------------------------------------------------------
// out[n] = h[n] * dinv[n]^2   (self-loop message; also initializes out)
// ---------------------------------------------------------------------------
__global__ __launch_bounds__(256) void k_selfloop_init(const float* __restrict__ h,
                                                       const float* __restrict__ dinv,
                                                       float* __restrict__ out, int n_nodes) {
    int idx = blockIdx.x * 256 + threadIdx.x;
    if (idx >= n_nodes * 32) return;
    int n = idx >> 5, c4 = idx & 31;
    float di = dinv[n];
    float w  = di * di;
    float4 v = ((const float4*)(h + (size_t)n * DNODE))[c4];
    float4 o = {v.x * w, v.y * w, v.z * w, v.w * w};
    ((float4*)(out + (size_t)n * DNODE))[c4] = o;
}

// ---------------------------------------------------------------------------
// edge scatter: out[dst] += h[src] * dinv[src]*dinv[dst]
// one wave per edge; each lane handles 4 cols (float4 gather + 4 fp32 atomics)
// ---------------------------------------------------------------------------
__global__ __launch_bounds__(256) void k_scatter_edges(const long long* __restrict__ esrc,
                                                       const long long* __restrict__ edst,
                                                       const float* __restrict__ dinv,
                                                       const float* __restrict__ h,
                                                       float* __restrict__ out, int n_edges) {
    int e = blockIdx.x * 8 + (threadIdx.x >> 5);
    if (e >= n_edges) return;
    int lane = threadIdx.x & 31;
    int s = (int)esrc[e];
    int d = (int)edst[e];
    float w = dinv[s] * dinv[d];
    float4 v = ((const float4*)(h + (size_t)s * DNODE))[lane];
    float* od = out + (size_t)d * DNODE + lane * 4;
    atomic_add_f32(od + 0, v.x * w);
    atomic_add_f32(od + 1, v.y * w);
    atomic_add_f32(od + 2, v.z * w);
    atomic_add_f32(od + 3, v.w * w);
}

// ---------------------------------------------------------------------------
// out = (out + bias) * mask
// ---------------------------------------------------------------------------
__global__ __launch_bounds__(256) void k_finalize(float* __restrict__ out,
                                                  const float* __restrict__ bias,
                                                  const int* __restrict__ mask, int n_nodes) {
    int idx = blockIdx.x * 256 + threadIdx.x;
    if (idx >= n_nodes * 32) return;
    int n = idx >> 5, c4 = idx & 31;
    float m  = (float)mask[n];
    float4 b = ((const float4*)bias)[c4];
    float4 v = ((float4*)(out + (size_t)n * DNODE))[c4];
    float4 o = {(v.x + b.x) * m, (v.y + b.y) * m, (v.z + b.z) * m, (v.w + b.w) * m};
    ((float4*)(out + (size_t)n * DNODE))[c4] = o;
}

// ---------------------------------------------------------------------------
extern "C" void kernel_launch(void* const* d_in, const int* in_sizes, int n_in,
                              void* d_out, int out_size, void* d_ws, size_t ws_size,
                              hipStream_t stream) {
    const float*     x    = (const float*)d_in[0];      // [N,128]
    const float*     W    = (const float*)d_in[1];      // [128,128]
    const float*     bias = (const float*)d_in[2];      // [128]
    const long long* eidx = (const long long*)d_in[3];  // [2,E] int64
    const int*       mask = (const int*)d_in[4];        // [N]

    const int n_nodes = in_sizes[4];
    const int n_edges = in_sizes[3] / 2;
    const long long* esrc = eidx;
    const long long* edst = eidx + n_edges;

    // workspace layout: [dinv: N floats][Wp: 16384 floats][h: N*128 floats]
    float* deg = (float*)d_ws;  // becomes dinv in place
    size_t wp_off = (((size_t)n_nodes * sizeof(float)) + 255) & ~(size_t)255;
    float* Wp = (float*)((char*)d_ws + wp_off);
    size_t h_off = wp_off + (size_t)DNODE * DNODE * sizeof(float);
    float* h = (float*)((char*)d_ws + h_off);

    float* out = (float*)d_out;

    int blk_n   = (n_nodes + 255) / 256;
    int blk_e   = (n_edges + 255) / 256;
    int blk_nc  = (n_nodes * 32 + 255) / 256;   // one thread per float4 of [N,128]
    int blk_gem = (n_nodes + 63) / 64;          // 4 waves/block * 16 rows/wave
    int blk_sc  = (n_edges + 7) / 8;            // 8 edges (waves) per block
    int blk_w   = (DNODE * DNODE + 255) / 256;

    k_deg_init     <<<blk_n,   256, 0, stream>>>(deg, n_nodes);
    k_deg_count    <<<blk_e,   256, 0, stream>>>(edst, deg, n_edges);
    k_deg_to_dinv  <<<blk_n,   256, 0, stream>>>(deg, n_nodes);
    k_pack_w       <<<blk_w,   256, 0, stream>>>(W, Wp);
    k_gemm_wmma    <<<blk_gem, 128, 0, stream>>>(x, Wp, mask, h, n_nodes);
    k_selfloop_init<<<blk_nc,  256, 0, stream>>>(h, deg, out, n_nodes);
    k_scatter_edges<<<blk_sc,  256, 0, stream>>>(esrc, edst, deg, h, out, n_edges);
    k_finalize     <<<blk_nc,  256, 0, stream>>>(out, bias, mask, n_nodes);
}